// PaiConvISO_51402168599239
// MI455X (gfx1250) — compile-verified
//
#include <hip/hip_runtime.h>
#include <cstdint>
#include <cstddef>

// Problem constants (from the reference)
#define BATCH  4
#define NPTS   65536
#define KNBR   9
#define FIN    64
#define FOUT   64
#define TILE_N 16
#define NTHREADS 128   // 4 waves of 32

typedef float v2f __attribute__((ext_vector_type(2)));
typedef float v8f __attribute__((ext_vector_type(8)));

// Branchless ELU using the hardware transcendental unit:
//   x > 0 : max(x,0)=x,  exp2(0)-1 = 0        -> x
//   x <= 0: max(x,0)=0,  exp2(x*log2e)-1      -> e^x - 1
__device__ __forceinline__ float elu1(float v) {
    const float e = __builtin_amdgcn_exp2f(fminf(v, 0.0f) * 1.4426950408889634f) - 1.0f;
    return fmaxf(v, 0.0f) + e;
}

__global__ __launch_bounds__(NTHREADS)
void PaiConvISO_wmma_kernel(const float* __restrict__ x,      // [B][N][FIN]
                            const int*   __restrict__ nbr,    // [B][N][KNBR]
                            const float* __restrict__ w,      // [FOUT][FIN]
                            const float* __restrict__ bias,   // [FOUT]
                            const float* __restrict__ adjw,   // [N][KNBR][KNBR]
                            float*       __restrict__ out)    // [B][N][FOUT]
{
    // LDS staging
    __shared__ float sE[KNBR * FIN * TILE_N];          // elu(xw): [t][f][p], 36 KB
    __shared__ float sAdj[TILE_N * KNBR * KNBR];       // [p][k][t], 5.2 KB
    __shared__ int   sIdx[TILE_N * KNBR];              // neighbor indices
    __shared__ float sOut[TILE_N * FOUT];              // output transpose tile, 4 KB

    const int tid = threadIdx.x;
    const int n0  = blockIdx.x * TILE_N;
    const int b   = blockIdx.y;

    // Stage adjweight tile + neighbor indices (coalesced)
    for (int i = tid; i < TILE_N * KNBR * KNBR; i += NTHREADS) {
        const int p = i / (KNBR * KNBR);
        sAdj[i] = adjw[(size_t)(n0 + p) * (KNBR * KNBR) + (i % (KNBR * KNBR))];
    }
    for (int i = tid; i < TILE_N * KNBR; i += NTHREADS) {
        sIdx[i] = nbr[((size_t)b * NPTS + n0) * KNBR + i];
    }
    __syncthreads();

    // ---------------- Phase 1: gather + adjweight combine + elu ----------------
    // thread = (point p, feature-group g); handles features f = g*8 .. g*8+7
    {
        const int p = tid >> 3;
        const int g = tid & 7;

        float xv[KNBR][8];
        #pragma unroll
        for (int k = 0; k < KNBR; ++k) {
            const size_t row = ((size_t)b * NPTS + (size_t)sIdx[p * KNBR + k]) * FIN + g * 8;
            const float4 lo = *(const float4*)(x + row);
            const float4 hi = *(const float4*)(x + row + 4);
            xv[k][0] = lo.x; xv[k][1] = lo.y; xv[k][2] = lo.z; xv[k][3] = lo.w;
            xv[k][4] = hi.x; xv[k][5] = hi.y; xv[k][6] = hi.z; xv[k][7] = hi.w;
        }

        #pragma unroll
        for (int t = 0; t < KNBR; ++t) {
            float acc[8];
            #pragma unroll
            for (int j = 0; j < 8; ++j) acc[j] = 0.0f;
            #pragma unroll
            for (int k = 0; k < KNBR; ++k) {
                const float aw = sAdj[p * (KNBR * KNBR) + k * KNBR + t];
                #pragma unroll
                for (int j = 0; j < 8; ++j) acc[j] = fmaf(xv[k][j], aw, acc[j]);
            }
            #pragma unroll
            for (int j = 0; j < 8; ++j) {
                // [t][f][p] layout: B-operand fetch is conflict-free (row of 16 floats)
                sE[(t * FIN + (g * 8 + j)) * TILE_N + p] = elu1(acc[j]);
            }
        }
    }
    __syncthreads();

    // ---------------- Phase 2: 16x16x4 f32 WMMA GEMM, bias+elu, tap max ----------------
    const int wv   = tid >> 5;       // wave id 0..3  -> output-channel tile
    const int lane = tid & 31;
    const int half = lane >> 4;      // 0: lanes 0-15, 1: lanes 16-31
    const int lm   = lane & 15;
    const int o0   = wv * 16;

    // A-operand (conv_w) in ISA 16x4 layout: lane = row M (=o), VGPR0/1 = K, K+1
    // (K = 4*s + 2*half for step s). Preload all 16 k-steps: reused across 9 taps.
    float a0[16], a1[16];
    #pragma unroll
    for (int s = 0; s < 16; ++s) {
        const int f = 4 * s + 2 * half;
        a0[s] = w[(size_t)(o0 + lm) * FIN + f];
        a1[s] = w[(size_t)(o0 + lm) * FIN + f + 1];
    }

    // bias per D element: D vgpr v holds (o = o0 + v + 8*half, p = lm)
    float bv[8];
    #pragma unroll
    for (int v = 0; v < 8; ++v) bv[v] = bias[o0 + v + 8 * half];

    v8f m;
    #pragma unroll
    for (int v = 0; v < 8; ++v) m[v] = -3.402823466e38f;

    for (int t = 0; t < KNBR; ++t) {
        // Dual accumulators: break the D->C RAW chain (dep distance 2 on matrix pipe).
        // Bias folded into acc0's initial value.
        v8f acc0, acc1 = {};
        #pragma unroll
        for (int v = 0; v < 8; ++v) acc0[v] = bv[v];

        #pragma unroll
        for (int s = 0; s < 16; s += 2) {
            {
                const int f = 4 * s + 2 * half;
                v2f A  = { a0[s], a1[s] };
                v2f Bv = { sE[(t * FIN + f) * TILE_N + lm],
                           sE[(t * FIN + f + 1) * TILE_N + lm] };
                acc0 = __builtin_amdgcn_wmma_f32_16x16x4_f32(
                    false, A, false, Bv, (short)0, acc0, false, false);
            }
            {
                const int f = 4 * (s + 1) + 2 * half;
                v2f A  = { a0[s + 1], a1[s + 1] };
                v2f Bv = { sE[(t * FIN + f) * TILE_N + lm],
                           sE[(t * FIN + f + 1) * TILE_N + lm] };
                acc1 = __builtin_amdgcn_wmma_f32_16x16x4_f32(
                    false, A, false, Bv, (short)0, acc1, false, false);
            }
        }

        #pragma unroll
        for (int v = 0; v < 8; ++v) {
            const float h = elu1(acc0[v] + acc1[v]);
            m[v] = fmaxf(m[v], h);
        }
    }

    // Write D tile into LDS as [p][o] so global stores coalesce
    #pragma unroll
    for (int v = 0; v < 8; ++v)
        sOut[lm * FOUT + o0 + 8 * half + v] = m[v];
    __syncthreads();

    // ---------------- Coalesced store with zero-pad on last vertex ----------------
    {
        const int p = tid >> 3;
        const int g = tid & 7;
        const int n = n0 + p;
        float4 r0 = *(const float4*)(sOut + p * FOUT + g * 8);
        float4 r1 = *(const float4*)(sOut + p * FOUT + g * 8 + 4);
        if (n == NPTS - 1) {
            r0 = make_float4(0.f, 0.f, 0.f, 0.f);
            r1 = make_float4(0.f, 0.f, 0.f, 0.f);
        }
        float* dst = out + ((size_t)b * NPTS + n) * FOUT + g * 8;
        *(float4*)(dst)     = r0;
        *(float4*)(dst + 4) = r1;
    }
}

extern "C" void kernel_launch(void* const* d_in, const int* in_sizes, int n_in,
                              void* d_out, int out_size, void* d_ws, size_t ws_size,
                              hipStream_t stream) {
    // setup_inputs order: x, t_vertex (unused), neighbor_index, conv_w, conv_b, adjweight
    const float* x    = (const float*)d_in[0];
    const int*   nbr  = (const int*)  d_in[2];
    const float* w    = (const float*)d_in[3];
    const float* bias = (const float*)d_in[4];
    const float* adjw = (const float*)d_in[5];
    float* out = (float*)d_out;

    dim3 grid(NPTS / TILE_N, BATCH);
    PaiConvISO_wmma_kernel<<<grid, NTHREADS, 0, stream>>>(x, nbr, w, bias, adjw, out);
}